// GCNIIBackbone_56616258896398
// MI455X (gfx1250) — compile-verified
//
#include <hip/hip_runtime.h>

#define NN     100000
#define NE     1600000
#define HID    128
#define ALPHA  0.1f
#define LAMBDA 0.5f
#define LDS_STRIDE 132   // 128 + 4 pad: bank = (4*m + k) % 64 -> conflict-free b64 reads

typedef __attribute__((ext_vector_type(2))) float v2f;
typedef __attribute__((ext_vector_type(8))) float v8f;

// ---------------- degree / normalization ----------------

__global__ __launch_bounds__(256) void k_init_dis(float* __restrict__ dis) {
    int i = blockIdx.x * 256 + threadIdx.x;
    if (i < NN) dis[i] = 1.0f;                 // self-loop contributes 1 to deg
}

__global__ __launch_bounds__(256) void k_deg(const long long* __restrict__ col,
                                             float* __restrict__ dis) {
    int e = blockIdx.x * 256 + threadIdx.x;
    if (e < NE) atomicAdd(&dis[(int)col[e]], 1.0f);
}

__global__ __launch_bounds__(256) void k_rsqrt(float* __restrict__ dis) {
    int i = blockIdx.x * 256 + threadIdx.x;
    if (i < NN) dis[i] = rsqrtf(dis[i]);
}

// ---------------- WMMA 16x16 tile GEMM (K=128, f32, 16x16x4) ----------------
// As: 16 x 128 tile in LDS (stride LDS_STRIDE). W: 128x128 row-major in global.
// lane%16 = M index for A frag and N index for B frag; lane/16 selects K pair.
__device__ __forceinline__ v8f tile_gemm(const float* __restrict__ As,
                                         const float* __restrict__ W,
                                         int ncol, int m, int half) {
    v8f acc = {};
#pragma unroll 8
    for (int kc = 0; kc < HID / 4; ++kc) {
        int k0 = 4 * kc + 2 * half;
        v2f a, b;
        a.x = As[m * LDS_STRIDE + k0];
        a.y = As[m * LDS_STRIDE + k0 + 1];
        b.x = W[k0 * HID + ncol];
        b.y = W[(k0 + 1) * HID + ncol];
        acc = __builtin_amdgcn_wmma_f32_16x16x4_f32(false, a, false, b,
                                                    (short)0, acc, false, false);
    }
    return acc;
}

// ---------------- initial projection: out = X @ W0 + b0 ----------------

__global__ __launch_bounds__(256) void k_proj(const float* __restrict__ X,
                                              const float* __restrict__ W,
                                              const float* __restrict__ bvec,
                                              float* __restrict__ out) {
    __shared__ float As[16 * LDS_STRIDE];
    int r0  = blockIdx.x * 16;
    int tid = threadIdx.x;
    for (int i = tid; i < 16 * HID; i += 256) {
        int rr = i >> 7, cc = i & 127;
        As[rr * LDS_STRIDE + cc] = X[(r0 + rr) * HID + cc];
    }
    __syncthreads();
    int wave = tid >> 5, lane = tid & 31;
    int n = lane & 15, half = lane >> 4;
    int ncol = wave * 16 + n;
    v8f acc = tile_gemm(As, W, ncol, n, half);
    float bias = bvec[ncol];
#pragma unroll
    for (int v = 0; v < 8; ++v) {
        int m = v + 8 * half;                  // C/D VGPR layout
        out[(r0 + m) * HID + ncol] = acc[v] + bias;
    }
}

// ---------------- self-loop init: agg = dis^2 * xin ----------------

__global__ __launch_bounds__(256) void k_self_loop(const float* __restrict__ xin,
                                                   const float* __restrict__ dis,
                                                   float* __restrict__ agg) {
    int i = blockIdx.x * 256 + threadIdx.x;    // over NN * (HID/4) float4 units
    if (i >= NN * (HID / 4)) return;
    int node = i >> 5;
    float w = dis[node] * dis[node];
    float4 v = ((const float4*)xin)[i];
    float4 o;
    o.x = w * v.x; o.y = w * v.y; o.z = w * v.z; o.w = w * v.w;
    ((float4*)agg)[i] = o;
}

// ---------------- edge scatter: one wave per edge, float4 per lane ----------------

__global__ __launch_bounds__(256) void k_scatter(const long long* __restrict__ row,
                                                 const long long* __restrict__ col,
                                                 const float* __restrict__ dis,
                                                 const float* __restrict__ xin,
                                                 float* __restrict__ agg) {
    int e = blockIdx.x * 8 + (threadIdx.x >> 5);
    if (e >= NE) return;
    int lane = threadIdx.x & 31;
    int r = (int)row[e];
    int c = (int)col[e];
    float w = dis[r] * dis[c];
    float4 v = ((const float4*)(xin + r * HID))[lane];   // gather hits L2 (x fits in 192MB)
    float* dst = agg + c * HID + lane * 4;
    atomicAdd(dst + 0, w * v.x);
    atomicAdd(dst + 1, w * v.y);
    atomicAdd(dst + 2, w * v.z);
    atomicAdd(dst + 3, w * v.w);
}

// ---------------- fused layer: h = (1-a)agg + a*x ; relu(b*(hW+bias) + (1-b)h) ----

__global__ __launch_bounds__(256) void k_layer(const float* __restrict__ xin,
                                               const float* __restrict__ agg,
                                               const float* __restrict__ W,
                                               const float* __restrict__ bvec,
                                               float beta,
                                               float* __restrict__ xout) {
    __shared__ float Hs[16 * LDS_STRIDE];
    int r0  = blockIdx.x * 16;
    int tid = threadIdx.x;
    for (int i = tid; i < 16 * HID; i += 256) {
        int rr = i >> 7, cc = i & 127;
        int g = (r0 + rr) * HID + cc;
        Hs[rr * LDS_STRIDE + cc] = agg[g] * (1.0f - ALPHA) + xin[g] * ALPHA;
    }
    __syncthreads();
    int wave = tid >> 5, lane = tid & 31;
    int n = lane & 15, half = lane >> 4;
    int ncol = wave * 16 + n;
    v8f acc = tile_gemm(Hs, W, ncol, n, half);
    float bias = bvec[ncol];
#pragma unroll
    for (int v = 0; v < 8; ++v) {
        int m = v + 8 * half;
        float h   = Hs[m * LDS_STRIDE + ncol];
        float val = (acc[v] + bias) * beta + (1.0f - beta) * h;
        xout[(r0 + m) * HID + ncol] = fmaxf(val, 0.0f);
    }
}

// ---------------- driver ----------------

extern "C" void kernel_launch(void* const* d_in, const int* in_sizes, int n_in,
                              void* d_out, int out_size, void* d_ws, size_t ws_size,
                              hipStream_t stream) {
    (void)in_sizes; (void)n_in; (void)out_size; (void)ws_size;

    const float*     x    = (const float*)d_in[0];
    const long long* edge = (const long long*)d_in[1];   // int64 (2, NE)
    const float*     W0   = (const float*)d_in[2];
    const float*     b0   = (const float*)d_in[3];
    const float*     Ws   = (const float*)d_in[4];       // (8, 128, 128)
    const float*     bs   = (const float*)d_in[5];       // (8, 128)
    float*           out  = (float*)d_out;

    const long long* row = edge;
    const long long* col = edge + NE;

    // workspace: dis[NN] | bufA[NN*HID] | agg[NN*HID]
    char*  ws  = (char*)d_ws;
    float* dis = (float*)ws;
    size_t off = (((size_t)NN * 4) + 255) & ~(size_t)255;
    float* bufA = (float*)(ws + off);
    off += (((size_t)NN * HID * 4) + 255) & ~(size_t)255;
    float* agg  = (float*)(ws + off);

    // symmetric degree normalization (recomputed each launch; deterministic)
    k_init_dis<<<(NN + 255) / 256, 256, 0, stream>>>(dis);
    k_deg     <<<(NE + 255) / 256, 256, 0, stream>>>(col, dis);
    k_rsqrt   <<<(NN + 255) / 256, 256, 0, stream>>>(dis);

    // x0 = x @ W0 + b0  -> d_out (ping-pong start so layer 7 lands in d_out)
    k_proj<<<NN / 16, 256, 0, stream>>>(x, W0, b0, out);

    for (int i = 0; i < 8; ++i) {
        const float* xin  = (i & 1) ? bufA : out;
        float*       xout = (i & 1) ? out  : bufA;
        k_self_loop<<<(NN * (HID / 4) + 255) / 256, 256, 0, stream>>>(xin, dis, agg);
        k_scatter  <<<NE / 8, 256, 0, stream>>>(row, col, dis, xin, agg);
        float beta = LAMBDA / (float)(i + 1);
        k_layer<<<NN / 16, 256, 0, stream>>>(xin, agg, Ws + (size_t)i * HID * HID,
                                             bs + i * HID, beta, xout);
    }
}